// PointMLP_14766097564245
// MI455X (gfx1250) — compile-verified
//
#include <hip/hip_runtime.h>
#include <math.h>

// ---------------- constants ----------------
#define BB   8
#define NN   4096
#define SS   1024
#define KNN_ 24
#define EPSF 1e-5f

#define MCOLS   (BB*SS*KNN_)   // 196608
#define EMB_M   (BB*NN)        // 32768
#define QCOLS   (BB*SS)        // 8192
#define KPAD    144            // padded K for the 131-channel transfer GEMM
#define LDW_LDS 80             // LDS row stride (floats): 2*80 % 64 = 32 -> no bank clash

typedef __attribute__((ext_vector_type(2))) float v2f;
typedef __attribute__((ext_vector_type(8))) float v8f;

// =====================================================================
// WMMA f32 GEMM with block-cooperative async LDS staging of the B panel.
//   Y[o, m] = sum_k W[o,k] * X[k,m] + bias[o]
// Block = 256 threads = 8 waves; block computes a full 128(O) x 64(N) strip.
// B panel staged in 16-row K-chunks via global_load_async_to_lds_b128.
// Requires: O==128, K%16==0, M%64==0.
// =====================================================================
__global__ void __launch_bounds__(256)
k_gemm_wmma(const float* __restrict__ W, int ldw,
            const float* __restrict__ X, int ldx,
            const float* __restrict__ bias,
            float* __restrict__ Y, int ldy,
            int Kdim, int M) {
    __shared__ float sB[16 * LDW_LDS];   // 16 K-rows x 64 cols (stride 80)

    const int lane  = threadIdx.x & 31;
    const int wv    = threadIdx.x >> 5;     // 0..7 -> O tile
    const int col0  = blockIdx.x << 6;      // 64-column strip
    const int orow0 = wv << 4;

    const int arow  = lane & 15;            // A: M index within tile
    const int khalf = (lane >> 4) << 1;     // K sub-pair (0 or 2)
    const float* Wrow = W + (size_t)(orow0 + arow) * ldw + khalf;

    // staging coords (all 256 threads): one b128 (4 floats) each
    const int srow = threadIdx.x >> 4;          // 0..15
    const int scol = (threadIdx.x & 15) << 2;   // 0..60 step 4
    const unsigned ldsOff =
        (unsigned)(unsigned long long)(&sB[srow * LDW_LDS + scol]);

    const int bcol = lane & 15;             // B fragment column within tile

    v8f acc0 = {}, acc1 = {}, acc2 = {}, acc3 = {};

    for (int k0 = 0; k0 < Kdim; k0 += 16) {
        __syncthreads();   // previous chunk fully consumed
        {
            const float* gp = X + (size_t)(k0 + srow) * ldx + col0 + scol;
            asm volatile("global_load_async_to_lds_b128 %0, %1, off"
                         :: "v"(ldsOff), "v"(gp) : "memory");
            asm volatile("s_wait_asynccnt 0x0" ::: "memory");
        }
        __syncthreads();   // staged chunk visible to all waves

        #pragma unroll
        for (int kk = 0; kk < 16; kk += 4) {
            v2f a;
            a.x = Wrow[k0 + kk];
            a.y = Wrow[k0 + kk + 1];
            const float* l0 = &sB[(kk + khalf) * LDW_LDS + bcol];
            const float* l1 = l0 + LDW_LDS;
            v2f b0, b1, b2, b3;
            b0.x = l0[0];   b0.y = l1[0];
            b1.x = l0[16];  b1.y = l1[16];
            b2.x = l0[32];  b2.y = l1[32];
            b3.x = l0[48];  b3.y = l1[48];
            acc0 = __builtin_amdgcn_wmma_f32_16x16x4_f32(false, a, false, b0, (short)0, acc0, false, false);
            acc1 = __builtin_amdgcn_wmma_f32_16x16x4_f32(false, a, false, b1, (short)0, acc1, false, false);
            acc2 = __builtin_amdgcn_wmma_f32_16x16x4_f32(false, a, false, b2, (short)0, acc2, false, false);
            acc3 = __builtin_amdgcn_wmma_f32_16x16x4_f32(false, a, false, b3, (short)0, acc3, false, false);
        }
    }

    // C/D layout: lanes 0-15 -> N=lane, M=v ; lanes 16-31 -> N=lane-16, M=v+8
    const int ncol  = lane & 15;
    const int mbase = (lane >> 4) << 3;
    #pragma unroll
    for (int v = 0; v < 8; ++v) {
        const int o = orow0 + mbase + v;
        const float bi = bias ? bias[o] : 0.0f;
        float* yr = Y + (size_t)o * ldy + col0 + ncol;
        yr[0]  = acc0[v] + bi;
        yr[16] = acc1[v] + bi;
        yr[32] = acc2[v] + bi;
        yr[48] = acc3[v] + bi;
    }
}

// =====================================================================
// Elementwise / reduction helpers
// =====================================================================
__global__ void k_zero(float* p, int n) {
    int i = blockIdx.x * blockDim.x + threadIdx.x;
    if (i < n) p[i] = 0.0f;
}

// embedding: F0[c, b*N+n] = sum_d W[c,d]*pts[b,d,n] + bias[c]
__global__ void k_embed(const float* __restrict__ pts, const float* __restrict__ W,
                        const float* __restrict__ bias, float* __restrict__ F0) {
    int i = blockIdx.x * blockDim.x + threadIdx.x;
    if (i >= 64 * EMB_M) return;
    int c = i >> 15;            // / 32768
    int m = i & (EMB_M - 1);
    int b = m >> 12;            // / 4096
    int n = m & (NN - 1);
    const float* p = pts + (size_t)b * 3 * NN + n;
    F0[i] = W[c*3+0]*p[0] + W[c*3+1]*p[NN] + W[c*3+2]*p[2*NN] + bias[c];
}

// per-channel sum / sumsq over M columns of Y [O, ldy]
__global__ void k_chstats(const float* __restrict__ Y, int ldy, int M,
                          float* __restrict__ stats) {
    __shared__ float sh[256], sh2[256];
    const int o = blockIdx.y;
    const float* row = Y + (size_t)o * ldy;
    float s = 0.f, s2 = 0.f;
    for (int i = blockIdx.x * blockDim.x + threadIdx.x; i < M; i += gridDim.x * blockDim.x) {
        float v = row[i]; s += v; s2 += v * v;
    }
    sh[threadIdx.x] = s; sh2[threadIdx.x] = s2;
    __syncthreads();
    for (int off = 128; off > 0; off >>= 1) {
        if (threadIdx.x < off) {
            sh[threadIdx.x]  += sh[threadIdx.x + off];
            sh2[threadIdx.x] += sh2[threadIdx.x + off];
        }
        __syncthreads();
    }
    if (threadIdx.x == 0) {
        atomicAdd(&stats[2*o],   sh[0]);
        atomicAdd(&stats[2*o+1], sh2[0]);
    }
}

// BN finalize: scale = g * rsqrt(var+eps); shift = be - mu*scale  (biased var)
__global__ void k_bnfin(const float* __restrict__ stats, const float* __restrict__ g,
                        const float* __restrict__ be, float invM,
                        float* __restrict__ scale, float* __restrict__ shift, int O) {
    int o = blockIdx.x * blockDim.x + threadIdx.x;
    if (o >= O) return;
    float mu  = stats[2*o] * invM;
    float var = stats[2*o+1] * invM - mu * mu;
    float sc  = g[o] * rsqrtf(var + EPSF);
    scale[o] = sc;
    shift[o] = be[o] - mu * sc;
}

// y = scale*y + shift (+res) (relu), in place
__global__ void k_bnapply(float* __restrict__ Y, int ldy, int M,
                          const float* __restrict__ scale, const float* __restrict__ shift,
                          const float* __restrict__ res, int doRelu) {
    const int o = blockIdx.y;
    const int m = blockIdx.x * blockDim.x + threadIdx.x;
    if (m >= M) return;
    size_t off = (size_t)o * ldy + m;
    float v = scale[o] * Y[off] + shift[o];
    if (res) v += res[off];
    if (doRelu) v = fmaxf(v, 0.0f);
    Y[off] = v;
}

// =====================================================================
// FPS: one workgroup (256 thr) per batch; dmin in LDS; 1024 sequential iters
// =====================================================================
__global__ void k_fps(const float* __restrict__ pts, int* __restrict__ fpsIdx) {
    __shared__ float dmin[NN];
    __shared__ float bv[256];
    __shared__ int   bi[256];
    __shared__ int   curFar;
    const int b = blockIdx.x, t = threadIdx.x;
    const float* px = pts + (size_t)b * 3 * NN;
    for (int n = t; n < NN; n += 256) dmin[n] = 1e10f;
    if (t == 0) curFar = 0;
    __syncthreads();
    for (int it = 0; it < SS; ++it) {
        const int far = curFar;
        if (t == 0) fpsIdx[b * SS + it] = far;   // emits incoming carry (first is 0)
        const float cx = px[far], cy = px[NN + far], cz = px[2*NN + far];
        float bestv = -1e30f; int besti = 0x7fffffff;
        for (int n = t; n < NN; n += 256) {
            float dx = px[n] - cx, dy = px[NN+n] - cy, dz = px[2*NN+n] - cz;
            float d  = dx*dx + dy*dy + dz*dz;
            float dm = fminf(dmin[n], d);
            dmin[n] = dm;
            if (dm > bestv || (dm == bestv && n < besti)) { bestv = dm; besti = n; }
        }
        bv[t] = bestv; bi[t] = besti;
        __syncthreads();
        for (int off = 128; off > 0; off >>= 1) {
            if (t < off) {
                float v2 = bv[t+off]; int i2 = bi[t+off];
                if (v2 > bv[t] || (v2 == bv[t] && i2 < bi[t])) { bv[t] = v2; bi[t] = i2; }
            }
            __syncthreads();
        }
        if (t == 0) curFar = bi[0];
        __syncthreads();
    }
}

// =====================================================================
// KNN: one workgroup (128 thr) per (b,s); distances in LDS; 24 argmin passes
// =====================================================================
__global__ void k_knn(const float* __restrict__ pts, const int* __restrict__ fpsIdx,
                      int* __restrict__ knnIdx) {
    __shared__ float dist[NN];
    __shared__ float bv[128];
    __shared__ int   bi[128];
    const int bs = blockIdx.x;
    const int b  = bs >> 10;
    const int t  = threadIdx.x;
    const float* px = pts + (size_t)b * 3 * NN;
    const int ci = fpsIdx[bs];
    const float cx = px[ci], cy = px[NN + ci], cz = px[2*NN + ci];
    const float cn = cx*cx + cy*cy + cz*cz;
    for (int n = t; n < NN; n += 128) {
        float x = px[n], y = px[NN+n], z = px[2*NN+n];
        dist[n] = cn + (x*x + y*y + z*z) - 2.0f * (cx*x + cy*y + cz*z);
    }
    __syncthreads();
    for (int k = 0; k < KNN_; ++k) {
        float bestv = 3.5e38f; int besti = 0x7fffffff;
        for (int n = t; n < NN; n += 128) {
            float d = dist[n];
            if (d < bestv || (d == bestv && n < besti)) { bestv = d; besti = n; }
        }
        bv[t] = bestv; bi[t] = besti;
        __syncthreads();
        for (int off = 64; off > 0; off >>= 1) {
            if (t < off) {
                float v2 = bv[t+off]; int i2 = bi[t+off];
                if (v2 < bv[t] || (v2 == bv[t] && i2 < bi[t])) { bv[t] = v2; bi[t] = i2; }
            }
            __syncthreads();
        }
        if (t == 0) { knnIdx[(size_t)bs * KNN_ + k] = bi[0]; dist[bi[0]] = 3.4e38f; }
        __syncthreads();
    }
}

// =====================================================================
// Grouping: per-batch std (ddof=1) of (g_pts - mean) over S*K*67 elements
// =====================================================================
#define PERB (SS*KNN_*67)   // 1,646,592

__global__ void k_groupstats(const float* __restrict__ pts, const float* __restrict__ F,
                             const int* __restrict__ fpsIdx, const int* __restrict__ knnIdx,
                             float* __restrict__ bstats) {
    __shared__ float sh[256], sh2[256];
    const int b = blockIdx.y, t = threadIdx.x;
    float s = 0.f, s2 = 0.f;
    for (int i = blockIdx.x * blockDim.x + t; i < PERB; i += gridDim.x * blockDim.x) {
        int c = i % 67;
        int r = i / 67;
        int sp = r / KNN_;
        int k  = r % KNN_;
        int idx = knnIdx[((size_t)b * SS + sp) * KNN_ + k];
        int fi  = fpsIdx[b * SS + sp];
        float gv, mv;
        if (c < 64) {
            gv = F[(size_t)c * EMB_M + b * NN + idx];
            mv = F[(size_t)c * EMB_M + b * NN + fi];
        } else {
            int d = c - 64;
            gv = pts[((size_t)b * 3 + d) * NN + idx];
            mv = pts[((size_t)b * 3 + d) * NN + fi];
        }
        float diff = gv - mv;
        s += diff; s2 += diff * diff;
    }
    sh[t] = s; sh2[t] = s2;
    __syncthreads();
    for (int off = 128; off > 0; off >>= 1) {
        if (t < off) { sh[t] += sh[t+off]; sh2[t] += sh2[t+off]; }
        __syncthreads();
    }
    if (t == 0) {
        atomicAdd(&bstats[2*b],   sh[0]);
        atomicAdd(&bstats[2*b+1], sh2[0]);
    }
}

__global__ void k_bstd(const float* __restrict__ bstats, float* __restrict__ bstd) {
    int b = threadIdx.x;
    if (b >= BB) return;
    double n  = (double)PERB;
    double s  = (double)bstats[2*b];
    double s2 = (double)bstats[2*b+1];
    double var = (s2 - s * s / n) / (n - 1.0);
    float  std = (float)sqrt(var);
    bstd[b] = 1.0f / (std + EPSF);
}

// pad W_tr [128,131] -> Wp [128,KPAD] (zero cols 131..)
__global__ void k_padw(const float* __restrict__ W, float* __restrict__ Wp) {
    int i = blockIdx.x * blockDim.x + threadIdx.x;
    if (i >= 128 * KPAD) return;
    int o = i / KPAD, k = i % KPAD;
    Wp[i] = (k < 131) ? W[o * 131 + k] : 0.0f;
}

// build X [KPAD, MCOLS]: rows 0..66 normalized group feats, 67..130 anchor feats, rest zero
__global__ void k_buildx(const float* __restrict__ pts, const float* __restrict__ F,
                         const int* __restrict__ fpsIdx, const int* __restrict__ knnIdx,
                         const float* __restrict__ alpha, const float* __restrict__ beta,
                         const float* __restrict__ bstd, float* __restrict__ X) {
    int gid = blockIdx.x * blockDim.x + threadIdx.x;
    if (gid >= KPAD * MCOLS) return;
    int c = gid / MCOLS;
    int m = gid % MCOLS;
    int b = m / (SS * KNN_);
    int r = m % (SS * KNN_);
    int sp = r / KNN_;
    float v;
    if (c >= 131) {
        v = 0.0f;
    } else if (c >= 67) {
        int fi = fpsIdx[b * SS + sp];
        v = F[(size_t)(c - 67) * EMB_M + b * NN + fi];
    } else {
        int idx = knnIdx[m];
        int fi  = fpsIdx[b * SS + sp];
        float gv, mv;
        if (c < 64) {
            gv = F[(size_t)c * EMB_M + b * NN + idx];
            mv = F[(size_t)c * EMB_M + b * NN + fi];
        } else {
            int d = c - 64;
            gv = pts[((size_t)b * 3 + d) * NN + idx];
            mv = pts[((size_t)b * 3 + d) * NN + fi];
        }
        v = alpha[c] * ((gv - mv) * bstd[b]) + beta[c];
    }
    X[gid] = v;
}

// max over K=24 per (channel, b*S+s)
__global__ void k_maxpool(const float* __restrict__ Y, float* __restrict__ Z) {
    int i = blockIdx.x * blockDim.x + threadIdx.x;
    if (i >= 128 * QCOLS) return;
    int o   = i >> 13;
    int col = i & (QCOLS - 1);
    const float* p = Y + (size_t)o * MCOLS + (size_t)col * KNN_;
    float mx = p[0];
    #pragma unroll
    for (int k = 1; k < KNN_; ++k) mx = fmaxf(mx, p[k]);
    Z[i] = mx;
}

// out[b,o,s] <- G[o, b*S+s]
__global__ void k_writeout(const float* __restrict__ G, float* __restrict__ out) {
    int i = blockIdx.x * blockDim.x + threadIdx.x;
    if (i >= BB * 128 * SS) return;
    int b = i / (128 * SS);
    int r = i % (128 * SS);
    int o = r / SS;
    int s = r % SS;
    out[i] = G[(size_t)o * QCOLS + b * SS + s];
}

// =====================================================================
// Host side
// =====================================================================
static inline float* wsF(char*& p, size_t nFloats) {
    float* r = (float*)p;
    p += ((nFloats * sizeof(float)) + 255) & ~(size_t)255;
    return r;
}
static inline int* wsI(char*& p, size_t nInts) {
    int* r = (int*)p;
    p += ((nInts * sizeof(int)) + 255) & ~(size_t)255;
    return r;
}

extern "C" void kernel_launch(void* const* d_in, const int* in_sizes, int n_in,
                              void* d_out, int out_size, void* d_ws, size_t ws_size,
                              hipStream_t stream) {
    (void)in_sizes; (void)n_in; (void)out_size; (void)ws_size;
    const float* pts    = (const float*)d_in[0];
    const float* W_emb  = (const float*)d_in[1];
    const float* b_emb  = (const float*)d_in[2];
    const float* g_emb  = (const float*)d_in[3];
    const float* be_emb = (const float*)d_in[4];
    const float* alpha  = (const float*)d_in[5];
    const float* beta   = (const float*)d_in[6];
    const float* W_tr   = (const float*)d_in[7];
    const float* b_tr   = (const float*)d_in[8];
    const float* g_tr   = (const float*)d_in[9];
    const float* be_tr  = (const float*)d_in[10];
    const float* W_p1   = (const float*)d_in[11];
    const float* b_p1   = (const float*)d_in[12];
    const float* g_p1   = (const float*)d_in[13];
    const float* be_p1  = (const float*)d_in[14];
    const float* W_p2   = (const float*)d_in[15];
    const float* b_p2   = (const float*)d_in[16];
    const float* g_p2   = (const float*)d_in[17];
    const float* be_p2  = (const float*)d_in[18];
    const float* W_q1   = (const float*)d_in[19];
    const float* b_q1   = (const float*)d_in[20];
    const float* g_q1   = (const float*)d_in[21];
    const float* be_q1  = (const float*)d_in[22];
    const float* W_q2   = (const float*)d_in[23];
    const float* b_q2   = (const float*)d_in[24];
    const float* g_q2   = (const float*)d_in[25];
    const float* be_q2  = (const float*)d_in[26];

    char* p = (char*)d_ws;
    float* F      = wsF(p, (size_t)64 * EMB_M);      // embedding features [64, B*N]
    float* stats  = wsF(p, 256);
    float* scaleB = wsF(p, 128);
    float* shiftB = wsF(p, 128);
    float* bstats = wsF(p, 16);
    float* bstd   = wsF(p, 8);
    int*   fpsIdx = wsI(p, BB * SS);
    int*   knnIdx = wsI(p, (size_t)MCOLS);
    float* Wp     = wsF(p, 128 * KPAD);
    float* Xbuf   = wsF(p, (size_t)KPAD * MCOLS);    // also reused as Y2
    float* Y1     = wsF(p, (size_t)128 * MCOLS);
    float* Hbuf   = wsF(p, (size_t)128 * MCOLS);
    float* Z      = wsF(p, (size_t)128 * QCOLS);
    float* G1     = wsF(p, (size_t)128 * QCOLS);
    float* G2     = wsF(p, (size_t)128 * QCOLS);
    float* Y2     = Xbuf;

    // ---- embedding conv + BN + ReLU ----
    k_embed<<<(64 * EMB_M) / 256, 256, 0, stream>>>(pts, W_emb, b_emb, F);
    k_zero<<<1, 256, 0, stream>>>(stats, 128);
    k_chstats<<<dim3(32, 64), 256, 0, stream>>>(F, EMB_M, EMB_M, stats);
    k_bnfin<<<1, 64, 0, stream>>>(stats, g_emb, be_emb, 1.0f / EMB_M, scaleB, shiftB, 64);
    k_bnapply<<<dim3(EMB_M / 256, 64), 256, 0, stream>>>(F, EMB_M, EMB_M, scaleB, shiftB, nullptr, 1);

    // ---- FPS + KNN ----
    k_fps<<<BB, 256, 0, stream>>>(pts, fpsIdx);
    k_knn<<<BB * SS, 128, 0, stream>>>(pts, fpsIdx, knnIdx);

    // ---- grouping: per-batch std, build X [KPAD, MCOLS] ----
    k_zero<<<1, 64, 0, stream>>>(bstats, 16);
    k_groupstats<<<dim3(256, BB), 256, 0, stream>>>(pts, F, fpsIdx, knnIdx, bstats);
    k_bstd<<<1, 8, 0, stream>>>(bstats, bstd);
    k_padw<<<(128 * KPAD + 255) / 256, 256, 0, stream>>>(W_tr, Wp);
    k_buildx<<<(KPAD * MCOLS + 255) / 256, 256, 0, stream>>>(pts, F, fpsIdx, knnIdx, alpha, beta, bstd, Xbuf);

    // ---- PreExtraction: transfer (131->128), K padded to 144 ----
    k_gemm_wmma<<<MCOLS / 64, 256, 0, stream>>>(Wp, KPAD, Xbuf, MCOLS, b_tr, Y1, MCOLS, KPAD, MCOLS);
    k_zero<<<1, 256, 0, stream>>>(stats, 256);
    k_chstats<<<dim3(128, 128), 256, 0, stream>>>(Y1, MCOLS, MCOLS, stats);
    k_bnfin<<<2, 64, 0, stream>>>(stats, g_tr, be_tr, 1.0f / MCOLS, scaleB, shiftB, 128);
    k_bnapply<<<dim3(MCOLS / 256, 128), 256, 0, stream>>>(Y1, MCOLS, MCOLS, scaleB, shiftB, nullptr, 1);

    // ---- PreExtraction resblock: p1 ----
    k_gemm_wmma<<<MCOLS / 64, 256, 0, stream>>>(W_p1, 128, Y1, MCOLS, b_p1, Hbuf, MCOLS, 128, MCOLS);
    k_zero<<<1, 256, 0, stream>>>(stats, 256);
    k_chstats<<<dim3(128, 128), 256, 0, stream>>>(Hbuf, MCOLS, MCOLS, stats);
    k_bnfin<<<2, 64, 0, stream>>>(stats, g_p1, be_p1, 1.0f / MCOLS, scaleB, shiftB, 128);
    k_bnapply<<<dim3(MCOLS / 256, 128), 256, 0, stream>>>(Hbuf, MCOLS, MCOLS, scaleB, shiftB, nullptr, 1);

    // ---- PreExtraction resblock: p2 + residual ----
    k_gemm_wmma<<<MCOLS / 64, 256, 0, stream>>>(W_p2, 128, Hbuf, MCOLS, b_p2, Y2, MCOLS, 128, MCOLS);
    k_zero<<<1, 256, 0, stream>>>(stats, 256);
    k_chstats<<<dim3(128, 128), 256, 0, stream>>>(Y2, MCOLS, MCOLS, stats);
    k_bnfin<<<2, 64, 0, stream>>>(stats, g_p2, be_p2, 1.0f / MCOLS, scaleB, shiftB, 128);
    k_bnapply<<<dim3(MCOLS / 256, 128), 256, 0, stream>>>(Y2, MCOLS, MCOLS, scaleB, shiftB, Y1, 1);

    // ---- max-pool over K ----
    k_maxpool<<<(128 * QCOLS) / 256, 256, 0, stream>>>(Y2, Z);

    // ---- PosExtraction resblock: q1 ----
    k_gemm_wmma<<<QCOLS / 64, 256, 0, stream>>>(W_q1, 128, Z, QCOLS, b_q1, G1, QCOLS, 128, QCOLS);
    k_zero<<<1, 256, 0, stream>>>(stats, 256);
    k_chstats<<<dim3(8, 128), 256, 0, stream>>>(G1, QCOLS, QCOLS, stats);
    k_bnfin<<<2, 64, 0, stream>>>(stats, g_q1, be_q1, 1.0f / QCOLS, scaleB, shiftB, 128);
    k_bnapply<<<dim3(QCOLS / 256, 128), 256, 0, stream>>>(G1, QCOLS, QCOLS, scaleB, shiftB, nullptr, 1);

    // ---- PosExtraction resblock: q2 + residual ----
    k_gemm_wmma<<<QCOLS / 64, 256, 0, stream>>>(W_q2, 128, G1, QCOLS, b_q2, G2, QCOLS, 128, QCOLS);
    k_zero<<<1, 256, 0, stream>>>(stats, 256);
    k_chstats<<<dim3(8, 128), 256, 0, stream>>>(G2, QCOLS, QCOLS, stats);
    k_bnfin<<<2, 64, 0, stream>>>(stats, g_q2, be_q2, 1.0f / QCOLS, scaleB, shiftB, 128);
    k_bnapply<<<dim3(QCOLS / 256, 128), 256, 0, stream>>>(G2, QCOLS, QCOLS, scaleB, shiftB, Z, 1);

    // ---- write output [B,128,S] ----
    k_writeout<<<(BB * 128 * SS) / 256, 256, 0, stream>>>(G2, (float*)d_out);
}